// MultiHeadAttention_50646254354439
// MI455X (gfx1250) — compile-verified
//
#include <hip/hip_runtime.h>
#include <hip/hip_bf16.h>
#include <math.h>

// ---------------------------------------------------------------------------
// MI455X / gfx1250 implementation of the deformable multi-head attention ref.
// fp32 end-to-end; big GEMMs on V_WMMA_F32_16X16X4_F32 (wave32 fragments),
// B tiles staged through the Tensor Data Mover (tensor_load_to_lds, TENSORcnt
// double-buffered) into LDS; A fragments direct from global with prefetch.
// ---------------------------------------------------------------------------

typedef float v2f __attribute__((ext_vector_type(2)));
typedef float v8f __attribute__((ext_vector_type(8)));
typedef unsigned int u32x4 __attribute__((ext_vector_type(4)));
typedef int i32x4 __attribute__((ext_vector_type(4)));
typedef int i32x8 __attribute__((ext_vector_type(8)));

#define HH    8      // heads == M
#define DDIM  1024   // input dim
#define EDIM  1024   // embed dim
#define VDIM  128
#define KD    128
#define SSEQ  10
#define KPTS  3
#define NROWS 10240  // B*S
#define BATCH 1024
#define Q2ROW 1280   // S*KD
#define PVROW 384    // V*K
#define KTILE 32     // K-rows per TDM B-tile chunk

// workspace offsets (floats); total = 64,143,360 floats = ~257 MB
#define OFF_Q2  ((size_t)0)          // 10,485,760  q2 permuted layout (8,1024,1280)
#define OFF_PK  ((size_t)10485760)   // 10,485,760  k proj (H, 10240, 128) == feats
#define OFF_PV  ((size_t)20971520)   // 31,457,280  v proj (H, 10240, 384)
#define OFF_DL  ((size_t)52428800)   //    491,520  deltas (8192, 60)
#define OFF_VGX ((size_t)52920320)   //    245,760  normalized grid x (8192,10,3)
#define OFF_VGY ((size_t)53166080)   //    245,760  normalized grid y
#define OFF_AT0 ((size_t)53411840)   //    245,760  attn row s=0 per (m,b)
#define OFF_HD  ((size_t)53657600)   // 10,485,760  head_flat (10240, 1024)

// ---------------------------------------------------------------------------
// TDM: DMA a 2D tile (rows x cols of fp32, row stride ldelem) from global
// memory into LDS at lds_byte_addr.  D# per CDNA5 ISA ch.8:
//   group0: count=1 | lds_addr | global_addr(57b) | type=2
//   group1: data_size=4B, tensor_dim0=ldelem, tensor_dim1=rows,
//           tile_dim0=cols, tile_dim1=rows, tensor_dim0_stride=ldelem
//   groups 2/3: zero (2D tensor)
// Tracked by TENSORcnt; pair with __builtin_amdgcn_s_wait_tensorcnt.
// ---------------------------------------------------------------------------
__device__ __forceinline__ void tdm_load_tile_f32(const float* gsrc,
                                                  unsigned lds_byte_addr,
                                                  int rows, int cols, int ldelem) {
  const unsigned long long ga = (unsigned long long)(uintptr_t)gsrc;
  u32x4 g0;
  g0.x = 1u;                                                   // count=1 (valid user D#)
  g0.y = lds_byte_addr;                                        // LDS dest (bytes)
  g0.z = (unsigned)ga;                                         // global_addr[31:0]
  g0.w = ((unsigned)(ga >> 32) & 0x01FFFFFFu) | 0x80000000u;   // addr[56:32] | type=2

  i32x8 g1;
  g1[0] = (int)(2u << 16);                                     // data_size = 4 bytes
  g1[1] = (int)(((unsigned)ldelem & 0xFFFFu) << 16);           // tensor_dim0[15:0]
  g1[2] = (int)((((unsigned)ldelem >> 16) & 0xFFFFu) |
                (((unsigned)rows & 0xFFFFu) << 16));           // dim0[31:16] | dim1[15:0]
  g1[3] = (int)((((unsigned)rows >> 16) & 0xFFFFu) |
                (((unsigned)cols & 0xFFFFu) << 16));           // dim1[31:16] | tile_dim0
  g1[4] = rows & 0xFFFF;                                       // tile_dim1 (tile_dim2=0)
  g1[5] = ldelem;                                              // tensor_dim0_stride[31:0]
  g1[6] = 0;
  g1[7] = 0;

  i32x4 g2; g2[0] = 0; g2[1] = 0; g2[2] = 0; g2[3] = 0;
  i32x4 g3 = g2;
#if defined(__clang_major__) && (__clang_major__ >= 23)
  i32x8 g23;
  g23[0]=0; g23[1]=0; g23[2]=0; g23[3]=0; g23[4]=0; g23[5]=0; g23[6]=0; g23[7]=0;
  __builtin_amdgcn_tensor_load_to_lds(g0, g1, g2, g3, g23, 0);
#else
  __builtin_amdgcn_tensor_load_to_lds(g0, g1, g2, g3, 0);
#endif
}

// ---------------------------------------------------------------------------
// Wave-level 64x64 fp32 WMMA tile: C[64x64] += A[64xK] * B[Kx64].
// A fragment (16x4 f32): lanes 0-15 hold rows r, K=k0..k0+1; lanes 16-31 hold
// K=k0+2..k0+3.  B tiles are TDM-staged in LDS (double buffered, KTILE x 64).
// ---------------------------------------------------------------------------
__device__ __forceinline__ void wave_gemm_64x64(const float* __restrict__ A, int lda,
                                                const float* __restrict__ B, int ldb,
                                                int Kdim, int lane, v8f acc[4][4],
                                                float* lbuf /* [2][KTILE*64] */) {
  const int r     = lane & 15;
  const int half2 = (lane >> 4) * 2;
  const float* Abase = A + (size_t)r * lda;
  const unsigned lds0 = (unsigned)(uintptr_t)lbuf;

  // prime chunk 0
  tdm_load_tile_f32(B, lds0, KTILE, 64, ldb);

  const int nch = Kdim / KTILE;
  for (int c = 0; c < nch; ++c) {
    if (c + 1 < nch) {
      tdm_load_tile_f32(B + (size_t)(c + 1) * KTILE * ldb,
                        lds0 + (unsigned)(((c + 1) & 1) * (KTILE * 64 * 4)),
                        KTILE, 64, ldb);
      __builtin_amdgcn_s_wait_tensorcnt(1);   // chunk c landed in LDS
    } else {
      __builtin_amdgcn_s_wait_tensorcnt(0);
    }
    const float* lb = lbuf + (c & 1) * (KTILE * 64);

    for (int kk = 0; kk < KTILE; kk += 4) {
      const int kg = c * KTILE + kk + half2;   // global K of this lane's pair
      v2f a[4];
#pragma unroll
      for (int i = 0; i < 4; ++i)
        a[i] = *(const v2f*)(Abase + (size_t)(i * 16) * lda + kg);
      __builtin_prefetch(Abase + kg + 64, 0, 1);   // global_prefetch_b8
      const int kl = kk + half2;               // local K row in LDS tile
      v2f b[4];
#pragma unroll
      for (int t = 0; t < 4; ++t) {
        const int col = t * 16 + r;
        b[t].x = lb[kl * 64 + col];
        b[t].y = lb[(kl + 1) * 64 + col];
      }
#pragma unroll
      for (int i = 0; i < 4; ++i)
#pragma unroll
        for (int t = 0; t < 4; ++t)
          acc[i][t] = __builtin_amdgcn_wmma_f32_16x16x4_f32(
              false, a[i], false, b[t], (short)0, acc[i][t], false, false);
    }
    // all ds_loads of this buffer must land before TDM overwrites it (2 chunks on)
    asm volatile("s_wait_dscnt 0x0" ::: "memory");
  }
}

// ---------------------------------------------------------------------------
// Kernel 1: fused Q/K/V projection GEMM. One wave per 64x64 output tile.
//   A = hf [10240,1024] (q flattened), B = per-head weight columns.
//   640 cols/head: [0,128)=Wq -> scatter into q2 layout, [128,256)=Wk,
//   [256,640)=Wv.  Branch-free epilogue via (base, rowStride, colStride).
// ---------------------------------------------------------------------------
__global__ __launch_bounds__(32)
void proj_gemm_kernel(const float* __restrict__ hf,
                      const float* __restrict__ Wq,
                      const float* __restrict__ Wk,
                      const float* __restrict__ Wv,
                      float* __restrict__ q2,
                      float* __restrict__ pk,
                      float* __restrict__ pv) {
  __shared__ float lbuf[2][KTILE * 64];
  const int lane    = threadIdx.x & 31;
  const int rowBase = blockIdx.x * 64;
  const int colBase = blockIdx.y * 64;        // within 640 per-head columns
  const int h       = blockIdx.z;

  const float* Bmat; int ldb; int c0;
  float* dstBase; size_t rowStride; size_t colStride;
  if (colBase < 128) {
    Bmat = Wq + (size_t)h * DDIM * KD;    ldb = KD;    c0 = colBase;
    // q2 flat = (row*KD + c)*HH + h  (torch permute(1,2,3,0) reinterpretation)
    dstBase = q2 + h; rowStride = (size_t)KD * HH; colStride = HH;
  } else if (colBase < 256) {
    Bmat = Wk + (size_t)h * DDIM * KD;    ldb = KD;    c0 = colBase - 128;
    dstBase = pk + (size_t)h * NROWS * KD; rowStride = KD; colStride = 1;
  } else {
    Bmat = Wv + (size_t)h * DDIM * PVROW; ldb = PVROW; c0 = colBase - 256;
    dstBase = pv + (size_t)h * NROWS * PVROW; rowStride = PVROW; colStride = 1;
  }

  v8f acc[4][4];
#pragma unroll
  for (int i = 0; i < 4; ++i)
#pragma unroll
    for (int t = 0; t < 4; ++t)
#pragma unroll
      for (int g = 0; g < 8; ++g) acc[i][t][g] = 0.0f;

  wave_gemm_64x64(hf + (size_t)rowBase * DDIM, DDIM, Bmat + c0, ldb, DDIM, lane,
                  acc, &lbuf[0][0]);

  const int r = lane & 15;
  const int mhi = (lane >> 4) * 8;
#pragma unroll
  for (int i = 0; i < 4; ++i) {
#pragma unroll
    for (int t = 0; t < 4; ++t) {
#pragma unroll
      for (int g = 0; g < 8; ++g) {
        const int row = rowBase + i * 16 + g + mhi;
        const int c   = c0 + t * 16 + r;
        dstBase[(size_t)row * rowStride + (size_t)c * colStride] = acc[i][t][g];
      }
    }
  }
}

// ---------------------------------------------------------------------------
// Kernel 2: deltas = q2 @ dw.T + db   (8192 x 1280 x 60)
// ---------------------------------------------------------------------------
__global__ __launch_bounds__(64)
void delta_kernel(const float* __restrict__ q2, const float* __restrict__ dw,
                  const float* __restrict__ db, float* __restrict__ dl) {
  __shared__ float qrow[Q2ROW];
  const int r = blockIdx.x;
  for (int i = threadIdx.x; i < Q2ROW; i += 64)
    qrow[i] = q2[(size_t)r * Q2ROW + i];
  __syncthreads();
  const int j = threadIdx.x;
  if (j < 60) {
    float acc = 0.0f;
    const float* wrow = dw + (size_t)j * Q2ROW;
    for (int i = 0; i < Q2ROW; ++i) acc = fmaf(qrow[i], wrow[i], acc);
    dl[(size_t)r * 60 + j] = acc + db[j];
  }
}

// ---------------------------------------------------------------------------
// Kernel 3: pd = ref + deltas, then F.normalize over the y axis (dim 1).
// grid = (n2, x, s); 48 active threads = 16 y * 3 k.
// ---------------------------------------------------------------------------
__global__ __launch_bounds__(64)
void gridcoord_kernel(const float* __restrict__ dl,
                      float* __restrict__ vgx, float* __restrict__ vgy) {
  __shared__ float sx[3][16], sy[3][16];
  const int n2 = blockIdx.x, x = blockIdx.y, s = blockIdx.z;
  const int tid = threadIdx.x;
  const int y = tid & 15, kk = tid >> 4;
  float px = 0.0f, py = 0.0f;
  if (kk < 3) {
    const int flat = (n2 * 16 + y) * 16 + x;
    const size_t base = (size_t)flat * 60 + s * 6 + kk * 2;
    px = (float)x * (1.0f / 15.0f) + dl[base + 0];
    py = (float)y * (1.0f / 15.0f) + dl[base + 1];
    sx[kk][y] = px * px;
    sy[kk][y] = py * py;
  }
  __syncthreads();
  if (kk < 3) {
    float nx = 0.0f, ny = 0.0f;
    for (int yy = 0; yy < 16; ++yy) { nx += sx[kk][yy]; ny += sy[kk][yy]; }
    nx = fmaxf(sqrtf(nx), 1e-12f);
    ny = fmaxf(sqrtf(ny), 1e-12f);
    const int flat = (n2 * 16 + y) * 16 + x;
    const size_t o = (size_t)flat * 30 + s * 3 + kk;
    vgx[o] = px / nx;
    vgy[o] = py / ny;
  }
}

// ---------------------------------------------------------------------------
// Kernel 4: per-(m,b) fused: bilinear-sample 30x128 k-tile from pk, qk GEMM,
// softmax, attn_map row (s=0), head = attn @ vp into head_flat layout.
// ---------------------------------------------------------------------------
__global__ __launch_bounds__(128)
void attn_kernel(const float* __restrict__ q2, const float* __restrict__ pk,
                 const float* __restrict__ pv, const float* __restrict__ vgx,
                 const float* __restrict__ vgy, float* __restrict__ at0,
                 float* __restrict__ hd) {
  __shared__ float qs[SSEQ * VDIM];   // 1280: q_attn tile
  __shared__ float kv[30 * VDIM];     // 3840: sampled k tile, later vp tile
  __shared__ float att[SSEQ][32];     // qk / softmax rows (30 used)

  const int blk = blockIdx.x;         // = m*1024 + b = flat pixel index
  const int m = blk >> 10, b = blk & 1023;
  const int tid = threadIdx.x;

  for (int i = tid; i < SSEQ * VDIM; i += 128)
    qs[i] = q2[(size_t)m * (BATCH * Q2ROW) + (size_t)b * Q2ROW + i];

  // bilinear grid-sample the k tile (zeros padding, align_corners=False)
  const int n2 = blk >> 8;
  const int v = tid;                  // 128 threads = channel dim
  for (int ks = 0; ks < 30; ++ks) {
    const int kpt = ks / SSEQ, s = ks % SSEQ;   // samp index = k*S + s
    const size_t go = (size_t)blk * 30 + s * 3 + kpt;
    const float gx = vgx[go], gy = vgy[go];
    const float ix = ((gx + 1.0f) * 16.0f - 1.0f) * 0.5f;
    const float iy = ((gy + 1.0f) * 16.0f - 1.0f) * 0.5f;
    const float x0 = floorf(ix), y0 = floorf(iy);
    const float wx1 = ix - x0, wx0 = 1.0f - wx1;
    const float wy1 = iy - y0, wy0 = 1.0f - wy1;
    auto corner = [&](float xc, float yc, float wgt) -> float {
      int xi = (int)xc; xi = xi < 0 ? 0 : (xi > 15 ? 15 : xi);
      int yi = (int)yc; yi = yi < 0 ? 0 : (yi > 15 ? 15 : yi);
      const bool valid = (xc >= 0.0f) && (xc <= 15.0f) && (yc >= 0.0f) && (yc <= 15.0f);
      const float f = pk[((((size_t)n2 * 16 + yi) * 16 + xi) * SSEQ + s) * VDIM + v];
      return valid ? f * wgt : 0.0f;
    };
    float acc = corner(x0,        y0,        wx0 * wy0)
              + corner(x0 + 1.0f, y0,        wx1 * wy0)
              + corner(x0,        y0 + 1.0f, wx0 * wy1)
              + corner(x0 + 1.0f, y0 + 1.0f, wx1 * wy1);
    kv[ks * VDIM + v] = acc;
  }
  __syncthreads();

  // qk = q_attn . k_attn^T * 1/sqrt(128)
  for (int t = tid; t < SSEQ * 30; t += 128) {
    const int s = t / 30, j = t % 30;
    float acc = 0.0f;
    for (int i = 0; i < VDIM; ++i) acc = fmaf(qs[s * VDIM + i], kv[j * VDIM + i], acc);
    att[s][j] = acc * 0.08838834764831845f;
  }
  __syncthreads();

  // softmax over 30
  if (tid < SSEQ) {
    float mx = -1e30f;
    for (int j = 0; j < 30; ++j) mx = fmaxf(mx, att[tid][j]);
    float sum = 0.0f;
    for (int j = 0; j < 30; ++j) { float e = __expf(att[tid][j] - mx); att[tid][j] = e; sum += e; }
    const float inv = 1.0f / sum;
    for (int j = 0; j < 30; ++j) att[tid][j] *= inv;
  }
  __syncthreads();

  if (tid < 30) at0[(size_t)blk * 30 + tid] = att[0][tid];

  // reuse kv for the vp tile (30x128 per (m,b), contiguous)
  for (int i = tid; i < 30 * VDIM; i += 128)
    kv[i] = pv[(size_t)m * (BATCH * (size_t)SSEQ * PVROW) + (size_t)b * (KPTS * SSEQ * VDIM) + i];
  __syncthreads();

  // head[s][v] = sum_j att[s][j] * vp[j][v]; store head_flat (b*10+s, m*128+v)
  for (int t = tid; t < SSEQ * VDIM; t += 128) {
    const int s = t >> 7, vv = t & 127;
    float acc = 0.0f;
    for (int j = 0; j < 30; ++j) acc = fmaf(att[s][j], kv[j * VDIM + vv], acc);
    hd[(((size_t)b * SSEQ + s) * HH + m) * VDIM + vv] = acc;
  }
}

// ---------------------------------------------------------------------------
// Kernel 5: out = head_flat @ Wo  (10240 x 1024 x 1024), WMMA f32, 64x64/wave.
// ---------------------------------------------------------------------------
__global__ __launch_bounds__(32)
void out_gemm_kernel(const float* __restrict__ hd, const float* __restrict__ wo,
                     float* __restrict__ out) {
  __shared__ float lbuf[2][KTILE * 64];
  const int lane    = threadIdx.x & 31;
  const int rowBase = blockIdx.x * 64;
  const int colBase = blockIdx.y * 64;

  v8f acc[4][4];
#pragma unroll
  for (int i = 0; i < 4; ++i)
#pragma unroll
    for (int t = 0; t < 4; ++t)
#pragma unroll
      for (int g = 0; g < 8; ++g) acc[i][t][g] = 0.0f;

  wave_gemm_64x64(hd + (size_t)rowBase * EDIM, EDIM, wo + colBase, EDIM, EDIM, lane,
                  acc, &lbuf[0][0]);

  const int r = lane & 15;
  const int mhi = (lane >> 4) * 8;
#pragma unroll
  for (int i = 0; i < 4; ++i) {
#pragma unroll
    for (int t = 0; t < 4; ++t) {
#pragma unroll
      for (int g = 0; g < 8; ++g) {
        const int row = rowBase + i * 16 + g + mhi;
        const int col = colBase + t * 16 + r;
        out[(size_t)row * EDIM + col] = acc[i][t][g];
      }
    }
  }
}

// ---------------------------------------------------------------------------
// Kernel 6: attn_map = mean over heads of attn[:, :, 0, :]
// ---------------------------------------------------------------------------
__global__ __launch_bounds__(256)
void attnmap_kernel(const float* __restrict__ at0, float* __restrict__ outmap) {
  const int t = blockIdx.x * 256 + threadIdx.x;
  if (t < BATCH * 30) {
    float s = 0.0f;
    for (int m = 0; m < HH; ++m) s += at0[(size_t)m * (BATCH * 30) + t];
    outmap[t] = s * 0.125f;
  }
}

// ---------------------------------------------------------------------------
extern "C" void kernel_launch(void* const* d_in, const int* in_sizes, int n_in,
                              void* d_out, int out_size, void* d_ws, size_t ws_size,
                              hipStream_t stream) {
  const float* q  = (const float*)d_in[0];
  const float* Wq = (const float*)d_in[1];
  const float* Wk = (const float*)d_in[2];
  const float* Wv = (const float*)d_in[3];
  const float* Wo = (const float*)d_in[4];
  const float* dw = (const float*)d_in[5];
  const float* db = (const float*)d_in[6];
  float* ws  = (float*)d_ws;
  float* out = (float*)d_out;

  // 1) projections: 160 row tiles (64) x 10 col groups (64) x 8 heads, 1 wave each
  proj_gemm_kernel<<<dim3(160, 10, 8), 32, 0, stream>>>(
      q, Wq, Wk, Wv, ws + OFF_Q2, ws + OFF_PK, ws + OFF_PV);
  // 2) deltas
  delta_kernel<<<8192, 64, 0, stream>>>(ws + OFF_Q2, dw, db, ws + OFF_DL);
  // 3) normalized grid coords
  gridcoord_kernel<<<dim3(32, 16, 10), 64, 0, stream>>>(
      ws + OFF_DL, ws + OFF_VGX, ws + OFF_VGY);
  // 4) fused sample + attention + head
  attn_kernel<<<8192, 128, 0, stream>>>(
      ws + OFF_Q2, ws + OFF_PK, ws + OFF_PV, ws + OFF_VGX, ws + OFF_VGY,
      ws + OFF_AT0, ws + OFF_HD);
  // 5) output projection: 160 row tiles x 16 col groups (64)
  out_gemm_kernel<<<dim3(160, 16), 32, 0, stream>>>(ws + OFF_HD, Wo, out);
  // 6) attn_map tail of d_out
  attnmap_kernel<<<120, 256, 0, stream>>>(ws + OFF_AT0, out + 10485760);
}